// GramEmbedding_6167573037095
// MI455X (gfx1250) — compile-verified
//
#include <hip/hip_runtime.h>
#include <cmath>

// Problem constants (from reference): F=1024, H=256, E=D=64, B=4096, N=1280
#define F_ 1024
#define H_ 256
#define E_ 64
#define D_ 64
#define B_ 4096
#define N_ 1280

typedef __attribute__((ext_vector_type(2))) float v2f;
typedef __attribute__((ext_vector_type(8))) float v8f;
typedef __attribute__((ext_vector_type(4))) int   i32x4;

// ---------------------------------------------------------------------------
// Hardware tanh: v_tanh_f32 on the gfx1250 TRANS unit. Never fall back to
// OCML (software tanh would ~20x the VALU cost of the dominant kernel).
// Inline-asm path appends v_nop to satisfy the TRANS RAW hazard rule.
// ---------------------------------------------------------------------------
__device__ __forceinline__ float fast_tanh(float x) {
#if __has_builtin(__builtin_amdgcn_tanhf)
    return __builtin_amdgcn_tanhf(x);
#else
    float y;
    asm volatile("v_tanh_f32 %0, %1\n\tv_nop\n\tv_nop" : "=v"(y) : "v"(x));
    return y;
#endif
}

// ---------------------------------------------------------------------------
// Async global->LDS copy (gfx1250, ASYNCcnt-tracked), with sync fallback.
// Builtin signature (from hipcc diagnostic): param 1 is a non-const
// addrspace(1) pointer to int32x4; param 2 is the LDS-side pointer.
// ---------------------------------------------------------------------------
#if __has_builtin(__builtin_amdgcn_global_load_async_to_lds_b128)
#define HAVE_ASYNC_LDS 1
__device__ __forceinline__ void async_copy16(const float* g, float* l) {
    __builtin_amdgcn_global_load_async_to_lds_b128(
        (__attribute__((address_space(1))) i32x4*)(float*)g,
        (__attribute__((address_space(3))) i32x4*)l,
        /*offset=*/0, /*cpol=*/0);
}
#endif

__device__ __forceinline__ void wait_async0() {
#if __has_builtin(__builtin_amdgcn_s_wait_asynccnt)
    __builtin_amdgcn_s_wait_asynccnt(0);
#else
    asm volatile("s_wait_asynccnt 0x0" ::: "memory");
#endif
}

// ---------------------------------------------------------------------------
// full = concat(feat_emb, hidden_emb)  -- (N_, E_) row-major, float4 copies
// ---------------------------------------------------------------------------
__global__ void build_full_kernel(const float* __restrict__ fe,
                                  const float* __restrict__ he,
                                  float* __restrict__ full) {
    int i = blockIdx.x * blockDim.x + threadIdx.x;   // float4 index
    int base = i * 4;
    if (base >= N_ * E_) return;
    float4 v;
    if (base < F_ * E_) v = *(const float4*)(fe + base);
    else                v = *(const float4*)(he + (base - F_ * E_));
    *(float4*)(full + base) = v;
}

// ---------------------------------------------------------------------------
// Generic f32 WMMA GEMM: C(MxN) = A(MxK) @ B(KxN), row-major.
// One wave32 computes one 16x16 tile of C via v_wmma_f32_16x16x4_f32.
// Optional epilogue: add bias per column (bias != null),
//                    divide by row_div per row (row_div != null).
// blockDim = (32, GEMM_WAVES)
// ---------------------------------------------------------------------------
#define GEMM_WAVES 4

__global__ void wmma_gemm_f32_kernel(const float* __restrict__ A,
                                     const float* __restrict__ B,
                                     float* __restrict__ C,
                                     int M, int N, int K,
                                     int lda, int ldb, int ldc,
                                     const float* __restrict__ bias,
                                     const float* __restrict__ row_div) {
    const int tilesN   = N >> 4;
    const int numTiles = (M >> 4) * tilesN;
    int tile = blockIdx.x * blockDim.y + threadIdx.y;     // uniform per wave
    if (tile >= numTiles) return;
    int tm = tile / tilesN;
    int tn = tile - tm * tilesN;

    int lane = threadIdx.x;          // 0..31
    int lo   = lane & 15;
    int hi   = lane >> 4;            // 0 or 1

    // A: lane holds row (tm*16+lo), K pair starting at k0 + 2*hi
    const float* Arow = A + (long)(tm * 16 + lo) * lda + 2 * hi;
    // B: lane holds col (tn*16+lo), rows k0 + 2*hi + {0,1}
    const float* Bcol = B + (long)(2 * hi) * ldb + tn * 16 + lo;

    v8f c = {};
    #pragma unroll 4
    for (int k0 = 0; k0 < K; k0 += 4) {
        // Stream-ahead prefetch on the A panel (global_prefetch_b8);
        // speculative OOB prefetches are silently dropped.
        if ((k0 & 63) == 0) __builtin_prefetch(Arow + k0 + 64, 0, 3);
        v2f a = *(const v2f*)(Arow + k0);        // 8B load (ka even, lda even)
        v2f b;
        b.x = Bcol[(long)k0 * ldb];
        b.y = Bcol[(long)k0 * ldb + ldb];
        c = __builtin_amdgcn_wmma_f32_16x16x4_f32(
                /*neg_a=*/false, a, /*neg_b=*/false, b,
                /*c_mod=*/(short)0, c, /*reuse_a=*/false, /*reuse_b=*/false);
    }

    float bcol = bias ? bias[tn * 16 + lo] : 0.0f;
    #pragma unroll
    for (int r = 0; r < 8; ++r) {
        int row = tm * 16 + r + 8 * hi;          // ISA C/D layout
        float v = c[r] + bcol;
        if (row_div) v /= row_div[row];
        C[(long)row * ldc + tn * 16 + lo] = v;
    }
}

// ---------------------------------------------------------------------------
// escore[f,n] = mask[f,n] ? exp( sum_d tanh(fp[f,d] + fpn[n,d]) * u[d] ) : 0
// (w_bias already folded into fp by the projection epilogue)
// Also accumulates rowsum[f] += sum_n escore[f,n] (atomic per block-row).
// Block = 16x16 threads, one (f,n) pair per thread; tiles staged in LDS
// via GLOBAL_LOAD_ASYNC_TO_LDS_B128 when available.
// LDS row stride = E_+4 floats (272B): 16B-aligned for B128 AND bank index
// (4*nl + d) mod 64 is conflict-free across the 16 n-lanes.
// ---------------------------------------------------------------------------
#define LPAD 4

__global__ void score_kernel(const float* __restrict__ fp,    // (F_, 64)
                             const float* __restrict__ fpn,   // (N_, 64)
                             const float* __restrict__ u,     // (64)
                             const unsigned char* __restrict__ mask, // (F_, N_)
                             float* __restrict__ escore,      // (F_, N_)
                             float* __restrict__ rowsum) {    // (F_)
    __shared__ __align__(16) float s_fp [16][E_ + LPAD];
    __shared__ __align__(16) float s_fpn[16][E_ + LPAD];
    __shared__ float s_u  [E_];
    __shared__ float s_red[16][17];

    const int f0 = blockIdx.y * 16;
    const int n0 = blockIdx.x * 16;
    const int tid = threadIdx.y * 16 + threadIdx.x;   // 0..255

    {   // cooperative tile load: one 16B chunk of each array per thread
        int r  = tid >> 4;          // 0..15
        int c4 = (tid & 15) * 4;    // 0..60
#ifdef HAVE_ASYNC_LDS
        async_copy16(fp  + (long)(f0 + r) * E_ + c4, &s_fp [r][c4]);
        async_copy16(fpn + (long)(n0 + r) * E_ + c4, &s_fpn[r][c4]);
        if (tid < E_) s_u[tid] = u[tid];
        wait_async0();
#else
        float4 vfp  = *(const float4*)(fp  + (long)(f0 + r) * E_ + c4);
        float4 vfpn = *(const float4*)(fpn + (long)(n0 + r) * E_ + c4);
        *(float4*)&s_fp [r][c4] = vfp;
        *(float4*)&s_fpn[r][c4] = vfpn;
        if (tid < E_) s_u[tid] = u[tid];
#endif
    }
    __syncthreads();

    const int fl = threadIdx.y;
    const int nl = threadIdx.x;
    float acc = 0.0f;
    #pragma unroll 8
    for (int d = 0; d < E_; ++d)
        acc += fast_tanh(s_fp[fl][d] + s_fpn[nl][d]) * s_u[d];

    const int f = f0 + fl;
    const int n = n0 + nl;
    float e = mask[(long)f * N_ + n] ? __expf(acc) : 0.0f;
    escore[(long)f * N_ + n] = e;

    s_red[fl][nl] = e;
    __syncthreads();
    if (nl == 0) {
        float s = 0.0f;
        #pragma unroll
        for (int j = 0; j < 16; ++j) s += s_red[fl][j];
        atomicAdd(&rowsum[f], s);   // global_atomic_add_f32
    }
}

// ---------------------------------------------------------------------------
// Launch
// ---------------------------------------------------------------------------
static inline void launch_gemm(const float* A, const float* B, float* C,
                               int M, int N, int K, int lda, int ldb, int ldc,
                               const float* bias, const float* row_div,
                               hipStream_t stream) {
    int numTiles = (M >> 4) * (N >> 4);
    int blocks = (numTiles + GEMM_WAVES - 1) / GEMM_WAVES;
    wmma_gemm_f32_kernel<<<blocks, dim3(32, GEMM_WAVES), 0, stream>>>(
        A, B, C, M, N, K, lda, ldb, ldc, bias, row_div);
}

extern "C" void kernel_launch(void* const* d_in, const int* in_sizes, int n_in,
                              void* d_out, int out_size, void* d_ws, size_t ws_size,
                              hipStream_t stream) {
    const float*         values     = (const float*)d_in[0];  // (B_, F_)
    const float*         feat_emb   = (const float*)d_in[1];  // (F_, E_)
    const float*         hidden_emb = (const float*)d_in[2];  // (H_, E_)
    const float*         w_kernel   = (const float*)d_in[3];  // (2E_, D_)
    const float*         w_bias     = (const float*)d_in[4];  // (D_)
    const float*         u_kernel   = (const float*)d_in[5];  // (D_)
    const unsigned char* mask       = (const unsigned char*)d_in[6]; // (F_, N_) bool
    float* out = (float*)d_out;                               // (B_, E_)

    // Workspace carve-up (all chunk sizes multiples of 4 floats -> 16B aligned)
    float* ws        = (float*)d_ws;
    float* full      = ws; ws += (size_t)N_ * E_;   // 81920
    float* feat_proj = ws; ws += (size_t)F_ * D_;   // 65536 (bias folded in)
    float* full_proj = ws; ws += (size_t)N_ * D_;   // 81920
    float* escore    = ws; ws += (size_t)F_ * N_;   // 1310720
    float* rowsum    = ws; ws += (size_t)F_;        // 1024
    float* context   = ws; ws += (size_t)F_ * E_;   // 65536

    // 1. full = concat(feat_emb, hidden_emb)
    build_full_kernel<<<(N_ * E_ / 4 + 255) / 256, 256, 0, stream>>>(
        feat_emb, hidden_emb, full);

    // 2. feat_proj = feat_emb @ Wa + w_bias   (Wa = w_kernel[:64])
    launch_gemm(feat_emb, w_kernel, feat_proj,
                F_, D_, E_, E_, D_, D_, w_bias, nullptr, stream);

    // 3. full_proj = full @ Wb                (Wb = w_kernel[64:])
    launch_gemm(full, w_kernel + (size_t)E_ * D_, full_proj,
                N_, D_, E_, E_, D_, D_, nullptr, nullptr, stream);

    // 4. rowsum = 0; escore + rowsum (hot kernel: 84M v_tanh_f32)
    (void)hipMemsetAsync(rowsum, 0, F_ * sizeof(float), stream);
    score_kernel<<<dim3(N_ / 16, F_ / 16), dim3(16, 16), 0, stream>>>(
        feat_proj, full_proj, u_kernel, mask, escore, rowsum);

    // 5. context = (escore @ full) / rowsum[:,None]
    launch_gemm(escore, full, context,
                F_, E_, N_, N_, E_, E_, nullptr, rowsum, stream);

    // 6. out = values @ context   (memory-bound: streams 16MB of `values`)
    launch_gemm(values, context, out,
                B_, E_, F_, F_, E_, E_, nullptr, nullptr, stream);
}